// DeformableTransformer_33234456936629
// MI455X (gfx1250) — compile-verified
//
#include <hip/hip_runtime.h>
#include <hip/hip_bf16.h>

#define D_MODEL  256
#define N_HEADS  8
#define N_LEVELS 3
#define N_POINTS 4
#define D_HEAD   32
#define BS_      8
#define NQ_      900
#define SUMHW    21504   // 128*128 + 64*64 + 32*32

typedef __attribute__((ext_vector_type(2))) float v2f;
typedef __attribute__((ext_vector_type(8))) float v8f;
typedef __attribute__((ext_vector_type(4))) int   v4i;

// --------------------------------------------------------------------------
// gfx1250 async global->LDS staging (ASYNCcnt-tracked), with fallback.
// Builtin prototype (from hipcc diagnostic): first param is v4i* in the
// global (AS1, printed as __device__) address space.
// --------------------------------------------------------------------------
#if __has_builtin(__builtin_amdgcn_global_load_async_to_lds_b128) && \
    __has_builtin(__builtin_amdgcn_s_wait_asynccnt)
#define USE_ASYNC_LDS 1
typedef __attribute__((address_space(1))) v4i gv4i_t;   // global int4
typedef __attribute__((address_space(3))) v4i lv4i_t;   // LDS int4
#else
#define USE_ASYNC_LDS 0
#endif

__device__ __forceinline__ void stage_b128(const float* g, float* l)
{
#if USE_ASYNC_LDS
    __builtin_amdgcn_global_load_async_to_lds_b128((gv4i_t*)g, (lv4i_t*)l, 0, 0);
#else
    *(float4*)l = *(const float4*)g;
#endif
}

__device__ __forceinline__ void wait_async_stage()
{
#if USE_ASYNC_LDS
    __builtin_amdgcn_s_wait_asynccnt(0);
#endif
}

// ---------------------------------------------------------------------------
// LDS-staged WMMA f32 GEMM:  C[M,N] = A[M,K] * W[K,N] + bias[N]
// Block = 8 waves; wave w computes M-tile (blockIdx.x*8+w), all waves share
// the 16*BN-column group blockIdx.y. W K-slabs (SLAB x 16*BN) are staged into
// LDS via async b128 loads, double buffered. K must be a multiple of SLAB.
//
// f32 WMMA lane layout (ISA 7.12.2):
//   A 16x4 : lanes 0-15 -> M=lane, v0=K0,v1=K1 ; lanes 16-31 -> M=lane-16, K2/K3
//   B 4x16 : lanes 0-15 -> N=lane, v0=K0,v1=K1 ; lanes 16-31 -> N=lane-16, K2/K3
//   C/D    : VGPR r, lanes 0-15 -> (M=r, N=lane) ; lanes 16-31 -> (M=r+8, N=lane-16)
// ---------------------------------------------------------------------------
#define SLAB    32
#define BSTRIDE 72   // padded floats per B-slab row (288B: 16B-aligned, bank-spread)

template <int BN>
__global__ __launch_bounds__(256)
void gemm_bias_wmma_lds(const float* __restrict__ A,
                        const float* __restrict__ W,
                        const float* __restrict__ bias,
                        float* __restrict__ C,
                        int M, int N, int K)
{
    __shared__ __align__(16) float Bs[2][SLAB * BSTRIDE];

    const int tid  = threadIdx.x;
    const int wave = tid >> 5;
    const int lane = tid & 31;
    const int lm   = lane & 15;
    const int kh   = lane >> 4;
    const int NCOL = 16 * BN;

    const int m_tile = blockIdx.x * 8 + wave;
    const int n0     = blockIdx.y * NCOL;
    const bool active = (m_tile * 16) < M;
    const int  arow   = active ? (m_tile * 16 + lm) : lm;  // clamp: safe dummy row
    const float* __restrict__ Arow = A + (size_t)arow * K;

    // Cooperative stage of W[k0..k0+SLAB) x [n0..n0+NCOL) into Bs[b]
    auto stage = [&](int b, int k0) {
        const int nchunk = SLAB * NCOL / 4;           // float4 chunks
        for (int idx = tid; idx < nchunk; idx += 256) {
            const int r  = idx / (NCOL / 4);
            const int c4 = (idx % (NCOL / 4)) * 4;
            stage_b128(W + (size_t)(k0 + r) * N + (n0 + c4),
                       &Bs[b][r * BSTRIDE + c4]);
        }
    };

    v8f c[BN] = {};
    int buf = 0;

    stage(0, 0);
    wait_async_stage();
    __syncthreads();

    for (int k0 = 0; k0 < K; k0 += SLAB) {
        if (k0 + SLAB < K) {
            stage(buf ^ 1, k0 + SLAB);                  // prefetch next slab (async)
            __builtin_prefetch(Arow + k0 + SLAB, 0, 0); // next A line -> global_prefetch
        }
        const float* __restrict__ Bbuf = &Bs[buf][0];
#pragma unroll
        for (int kk = 0; kk < SLAB; kk += 4) {
            v2f a;
            a.x = Arow[k0 + kk + 2 * kh];
            a.y = Arow[k0 + kk + 2 * kh + 1];
#pragma unroll
            for (int t = 0; t < BN; ++t) {
                v2f b;
                b.x = Bbuf[(kk + 2 * kh) * BSTRIDE + t * 16 + lm];
                b.y = Bbuf[(kk + 2 * kh + 1) * BSTRIDE + t * 16 + lm];
                c[t] = __builtin_amdgcn_wmma_f32_16x16x4_f32(
                    false, a, false, b, (short)0, c[t], false, false);
            }
        }
        wait_async_stage();   // next-slab LDS writes landed (this wave)
        __syncthreads();      // all waves done reading Bs[buf] / staging Bs[buf^1]
        buf ^= 1;
    }

    if (active) {
#pragma unroll
        for (int t = 0; t < BN; ++t) {
            const int n  = n0 + t * 16 + lm;
            const float bv = bias[n];
#pragma unroll
            for (int r = 0; r < 8; ++r) {
                const int m = m_tile * 16 + r + 8 * kh;
                C[(size_t)m * N + n] = c[t][r] + bv;
            }
        }
    }
}

// ---------------------------------------------------------------------------
// Plain wave-per-tile WMMA GEMM (used for the tiny attn projection, N=96).
// ---------------------------------------------------------------------------
template <int BN>
__global__ __launch_bounds__(32)
void gemm_bias_wmma_f32(const float* __restrict__ A,
                        const float* __restrict__ W,
                        const float* __restrict__ bias,
                        float* __restrict__ C,
                        int M, int N, int K)
{
    const int lane = threadIdx.x & 31;
    const int lm   = lane & 15;
    const int kh   = lane >> 4;
    const int m0   = blockIdx.x * 16;
    const int n0   = blockIdx.y * (16 * BN);

    const float* __restrict__ Arow = A + (size_t)(m0 + lm) * K;

    v8f c[BN] = {};
    for (int k0 = 0; k0 < K; k0 += 4) {
        v2f a;
        a.x = Arow[k0 + 2 * kh];
        a.y = Arow[k0 + 2 * kh + 1];
        const float* __restrict__ Wr = W + (size_t)(k0 + 2 * kh) * N;
#pragma unroll
        for (int t = 0; t < BN; ++t) {
            const int n = n0 + t * 16 + lm;
            v2f b;
            b.x = Wr[n];
            b.y = Wr[N + n];
            c[t] = __builtin_amdgcn_wmma_f32_16x16x4_f32(
                false, a, false, b, (short)0, c[t], false, false);
        }
    }
#pragma unroll
    for (int t = 0; t < BN; ++t) {
        const int n  = n0 + t * 16 + lm;
        const float bv = bias[n];
#pragma unroll
        for (int r = 0; r < 8; ++r) {
            const int m = m0 + r + 8 * kh;
            C[(size_t)m * N + n] = c[t][r] + bv;
        }
    }
}

// ---------------------------------------------------------------------------
// Softmax over the 12 (level*point) attention logits per (b, q, head).
// ---------------------------------------------------------------------------
__global__ __launch_bounds__(256)
void softmax12_kernel(float* __restrict__ attn, int total)
{
    const int i = blockIdx.x * blockDim.x + threadIdx.x;
    if (i >= total) return;
    float* p = attn + (size_t)i * (N_LEVELS * N_POINTS);
    float m = p[0];
#pragma unroll
    for (int j = 1; j < N_LEVELS * N_POINTS; ++j) m = fmaxf(m, p[j]);
    float e[N_LEVELS * N_POINTS];
    float s = 0.f;
#pragma unroll
    for (int j = 0; j < N_LEVELS * N_POINTS; ++j) { e[j] = __expf(p[j] - m); s += e[j]; }
    const float inv = 1.f / s;
#pragma unroll
    for (int j = 0; j < N_LEVELS * N_POINTS; ++j) p[j] = e[j] * inv;
}

// ---------------------------------------------------------------------------
// Deformable sampling: one wave per (b, q, head), lane = channel d (D_HEAD=32).
// value layout [b, s, h*32+d] -> each corner read is one coalesced 128B line,
// and the whole 176MB value tensor is L2-resident (192MB global L2).
// ---------------------------------------------------------------------------
__device__ __forceinline__ float msda_corner(const float* __restrict__ vb,
                                             int st, int Hl, int Wl,
                                             int xi, int yi, float wt)
{
    const bool valid = (xi >= 0) & (xi < Wl) & (yi >= 0) & (yi < Hl);
    const int xc = min(max(xi, 0), Wl - 1);
    const int yc = min(max(yi, 0), Hl - 1);
    const float g = vb[(size_t)(st + yc * Wl + xc) * D_MODEL];
    return valid ? wt * g : 0.f;
}

__global__ __launch_bounds__(256)
void msda_sample_kernel(const float* __restrict__ value,  // [BS, SUMHW, 256]
                        const float* __restrict__ ref,    // [BS, NQ, 3, 2]
                        const float* __restrict__ offs,   // [BS, NQ, 192]
                        const float* __restrict__ attnw,  // [BS, NQ, 96] softmaxed
                        float* __restrict__ acc)          // [BS, NQ, 256]
{
    const int bq = blockIdx.x;           // b*NQ + q
    const int b  = bq / NQ_;
    const int h  = threadIdx.x >> 5;
    const int d  = threadIdx.x & 31;

    const int Hs[N_LEVELS] = {128, 64, 32};
    const int Ws[N_LEVELS] = {128, 64, 32};
    const int St[N_LEVELS] = {0, 16384, 20480};

    const float* __restrict__ rp = ref   + (size_t)bq * (N_LEVELS * 2);
    const float* __restrict__ of = offs  + (size_t)bq * (N_HEADS * N_LEVELS * N_POINTS * 2)
                                         + h * (N_LEVELS * N_POINTS * 2);
    const float* __restrict__ aw = attnw + (size_t)bq * (N_HEADS * N_LEVELS * N_POINTS)
                                         + h * (N_LEVELS * N_POINTS);
    const float* __restrict__ vb = value + (size_t)b * SUMHW * D_MODEL + h * D_HEAD + d;

    float a = 0.f;
#pragma unroll
    for (int l = 0; l < N_LEVELS; ++l) {
        const int Hl = Hs[l], Wl = Ws[l], st = St[l];
        const float rx = rp[l * 2 + 0];
        const float ry = rp[l * 2 + 1];
#pragma unroll
        for (int p = 0; p < N_POINTS; ++p) {
            // (ref + off/norm)*dim - 0.5  ==  ref*dim + off - 0.5
            const float px = rx * (float)Wl + of[l * 8 + p * 2 + 0] - 0.5f;
            const float py = ry * (float)Hl + of[l * 8 + p * 2 + 1] - 0.5f;
            const float fx = floorf(px), fy = floorf(py);
            const float lx = px - fx,   ly = py - fy;
            const int   x0 = (int)fx,   y0 = (int)fy;
            const float w  = aw[l * 4 + p];
            a += msda_corner(vb, st, Hl, Wl, x0,     y0,     w * (1.f - lx) * (1.f - ly));
            a += msda_corner(vb, st, Hl, Wl, x0 + 1, y0,     w * lx        * (1.f - ly));
            a += msda_corner(vb, st, Hl, Wl, x0,     y0 + 1, w * (1.f - lx) * ly);
            a += msda_corner(vb, st, Hl, Wl, x0 + 1, y0 + 1, w * lx        * ly);
        }
    }
    acc[(size_t)bq * D_MODEL + h * D_HEAD + d] = a;
}

// ---------------------------------------------------------------------------
extern "C" void kernel_launch(void* const* d_in, const int* in_sizes, int n_in,
                              void* d_out, int out_size, void* d_ws, size_t ws_size,
                              hipStream_t stream)
{
    (void)in_sizes; (void)n_in; (void)out_size; (void)ws_size;

    const float* query  = (const float*)d_in[0];
    const float* refp   = (const float*)d_in[1];
    const float* inpf   = (const float*)d_in[2];
    // d_in[3] spatial_shapes, d_in[4] level_start_index: compile-time constants here
    const float* W_off  = (const float*)d_in[5];
    const float* b_off  = (const float*)d_in[6];
    const float* W_attn = (const float*)d_in[7];
    const float* b_attn = (const float*)d_in[8];
    const float* W_val  = (const float*)d_in[9];
    const float* b_val  = (const float*)d_in[10];
    const float* W_out  = (const float*)d_in[11];
    const float* b_out  = (const float*)d_in[12];
    float* out = (float*)d_out;

    const int M_val  = BS_ * SUMHW;                        // 172032
    const int M_q    = BS_ * NQ_;                          // 7200
    const int N_off  = N_HEADS * N_LEVELS * N_POINTS * 2;  // 192
    const int N_attn = N_HEADS * N_LEVELS * N_POINTS;      // 96

    // Workspace carve-out:
    char* ws = (char*)d_ws;
    float* v_val  = (float*)(ws);                                          // 176,160,768 B
    float* v_off  = (float*)(ws + 176160768ull);                           //   5,529,600 B
    float* v_attn = (float*)(ws + 176160768ull + 5529600ull);              //   2,764,800 B
    float* v_acc  = (float*)(ws + 176160768ull + 5529600ull + 2764800ull); //   7,372,800 B

    // 1) value = input_flatten @ W_val + b_val         [172032 x 256]
    gemm_bias_wmma_lds<4><<<dim3(M_val / 16 / 8, D_MODEL / 64), 256, 0, stream>>>(
        inpf, W_val, b_val, v_val, M_val, D_MODEL, D_MODEL);

    // 2) offsets = query @ W_off + b_off               [7200 x 192]
    gemm_bias_wmma_lds<4><<<dim3((M_q / 16 + 7) / 8, N_off / 64), 256, 0, stream>>>(
        query, W_off, b_off, v_off, M_q, N_off, D_MODEL);

    // 3) attn logits = query @ W_attn + b_attn         [7200 x 96]
    gemm_bias_wmma_f32<2><<<dim3(M_q / 16, N_attn / 32), 32, 0, stream>>>(
        query, W_attn, b_attn, v_attn, M_q, N_attn, D_MODEL);

    // 4) softmax over 12 per (b,q,h)
    {
        const int total = M_q * N_HEADS;  // 57600
        softmax12_kernel<<<(total + 255) / 256, 256, 0, stream>>>(v_attn, total);
    }

    // 5) deformable bilinear sampling + attention-weighted accumulate
    msda_sample_kernel<<<M_q, 256, 0, stream>>>(v_val, refp, v_off, v_attn, v_acc);

    // 6) out = acc @ W_out + b_out                     [7200 x 256]
    gemm_bias_wmma_lds<4><<<dim3((M_q / 16 + 7) / 8, D_MODEL / 64), 256, 0, stream>>>(
        v_acc, W_out, b_out, out, M_q, D_MODEL, D_MODEL);
}